// HeteroGLSTM_77567109366013
// MI455X (gfx1250) — compile-verified
//
#include <hip/hip_runtime.h>
#include <math.h>

// ---------------- constants ----------------
constexpr int kNA = 100000;   // divisible by 16
constexpr int kNB = 50000;    // divisible by 16
constexpr int kC  = 64;
constexpr int kFIN = 32;

typedef __attribute__((ext_vector_type(2))) float v2f;
typedef __attribute__((ext_vector_type(8))) float v8f;

// ---------------- degree count ----------------
__global__ void count_kernel(float* __restrict__ cnt, const int* __restrict__ dst, int E) {
    int e = blockIdx.x * blockDim.x + threadIdx.x;
    if (e < E) atomicAdd(&cnt[dst[e]], 1.0f);
}

// ---------------- edge gather + atomic scatter (segment-sum) ----------------
// 16 threads per edge, float4 per thread (64 channels). t and agg are L2-resident.
__global__ void scatter_kernel(float* __restrict__ agg, const float* __restrict__ t,
                               const int* __restrict__ src, const int* __restrict__ dst, int E) {
    int gid = blockIdx.x * blockDim.x + threadIdx.x;
    int e = gid >> 4;
    if (e >= E) return;
    int q = (gid & 15) * 4;
    const float4 v = *(const float4*)(t + (size_t)src[e] * kC + q);
    float* d = agg + (size_t)dst[e] * kC + q;
    atomicAdd(d + 0, v.x);
    atomicAdd(d + 1, v.y);
    atomicAdd(d + 2, v.z);
    atomicAdd(d + 3, v.w);
}

// ---------------- WMMA f32 GEMM: dst[nrows x 64] (+)= diag(1/max(cnt,1)) * A[nrows x K] @ W[K x 64] + bias ----------------
// block = 128 threads = 4 waves; wave w owns columns [16w, 16w+16); block b owns rows [16b, 16b+16).
__global__ __launch_bounds__(128) void gemm64_wmma(
    float* __restrict__ dst,
    const float* __restrict__ A,
    const float* __restrict__ cnt,   // per-row mean denominator, or nullptr
    const float* __restrict__ W,     // row-major [K][64]
    const float* __restrict__ bias,  // [64] or nullptr
    int nrows, int K, int accumulate)
{
    const int lane  = threadIdx.x & 31;
    const int wave  = threadIdx.x >> 5;
    const int row0  = blockIdx.x * 16;
    const int n0    = wave * 16;
    const int ml    = lane & 15;   // A row / B,C column within tile
    const int khalf = lane >> 4;   // K split across lane halves

    const int arow = row0 + ml;
    float scale = 1.0f;
    if (cnt) scale = 1.0f / fmaxf(cnt[arow], 1.0f);

    v8f c = {};
    if (accumulate) {
#pragma unroll
        for (int r = 0; r < 8; ++r)
            c[r] = dst[(size_t)(row0 + r + 8 * khalf) * kC + n0 + ml];
    }

    const float* Arow = A + (size_t)arow * K + 2 * khalf;
    const float* Wcol = W + n0 + ml;
    for (int k0 = 0; k0 < K; k0 += 4) {
        v2f a, b;
        a.x = Arow[k0 + 0] * scale;
        a.y = Arow[k0 + 1] * scale;
        const int kb = k0 + 2 * khalf;
        b.x = Wcol[(size_t)(kb + 0) * kC];
        b.y = Wcol[(size_t)(kb + 1) * kC];
        c = __builtin_amdgcn_wmma_f32_16x16x4_f32(
                /*neg_a=*/false, a, /*neg_b=*/false, b,
                /*c_mod=*/(short)0, c, /*reuse_a=*/false, /*reuse_b=*/false);
    }

    const float bv = bias ? bias[n0 + ml] : 0.0f;
#pragma unroll
    for (int r = 0; r < 8; ++r)
        dst[(size_t)(row0 + r + 8 * khalf) * kC + n0 + ml] = c[r] + bv;
}

// ---------------- final LSTM elementwise (adds deferred lin_b, activations, combine) ----------------
__global__ void lstm_kernel(float* __restrict__ h, float* __restrict__ cn,
                            const float* __restrict__ g0, const float* __restrict__ g1,
                            const float* __restrict__ g2, const float* __restrict__ g3,
                            const float* __restrict__ cprev,
                            const float* __restrict__ lin_b, int t, long long n)
{
    long long i = (long long)blockIdx.x * blockDim.x + threadIdx.x;
    if (i >= n) return;
    const int ch = (int)(i & (kC - 1));
    const float bi = lin_b[(0 * 2 + t) * kC + ch];
    const float bf = lin_b[(1 * 2 + t) * kC + ch];
    const float bo = lin_b[(2 * 2 + t) * kC + ch];
    const float bg = lin_b[(3 * 2 + t) * kC + ch];
    const float iv = 1.0f / (1.0f + __expf(-(g0[i] + bi)));
    const float fv = 1.0f / (1.0f + __expf(-(g1[i] + bf)));
    const float ov = 1.0f / (1.0f + __expf(-(g2[i] + bo)));
    const float gv = tanhf(g3[i] + bg);
    const float cv = fv * cprev[i] + iv * gv;
    cn[i] = cv;
    h[i]  = ov * tanhf(cv);
}

// ---------------- host-side helpers ----------------
static inline void launch_gemm(hipStream_t s, float* dst, const float* A, const float* cnt,
                               const float* W, const float* bias, int nrows, int K, int acc) {
    gemm64_wmma<<<nrows / 16, 128, 0, s>>>(dst, A, cnt, W, bias, nrows, K, acc);
}

extern "C" void kernel_launch(void* const* d_in, const int* in_sizes, int n_in,
                              void* d_out, int out_size, void* d_ws, size_t ws_size,
                              hipStream_t stream) {
    (void)n_in; (void)out_size; (void)ws_size;

    const float* x_a     = (const float*)d_in[0];
    const float* x_b     = (const float*)d_in[1];
    const float* c_a     = (const float*)d_in[2];
    const float* c_b     = (const float*)d_in[3];
    const float* lin_w   = (const float*)d_in[4];   // (4,2,32,64)
    const float* lin_b   = (const float*)d_in[5];   // (4,2,64)
    const float* conv_wl = (const float*)d_in[6];   // (4,2,3,64,64)
    const float* conv_wr = (const float*)d_in[7];   // (4,2,3,64,64)
    const float* conv_b  = (const float*)d_in[8];   // (4,2,3,64)
    const int* src_aa = (const int*)d_in[9];
    const int* dst_aa = (const int*)d_in[10];
    const int* src_ab = (const int*)d_in[11];
    const int* dst_ab = (const int*)d_in[12];
    const int* src_bb = (const int*)d_in[13];
    const int* dst_bb = (const int*)d_in[14];
    const int E_aa = in_sizes[9], E_ab = in_sizes[11], E_bb = in_sizes[13];

    const size_t na = (size_t)kNA * kC;
    const size_t nb = (size_t)kNB * kC;

    // workspace carve-out (floats)
    float* ws     = (float*)d_ws;
    float* gate_a = ws;  ws += 4 * na;
    float* gate_b = ws;  ws += 4 * nb;
    float* tmpA   = ws;  ws += 2 * na;     // ping/pong for a-features
    float* tmpB   = ws;  ws += 2 * nb;     // ping/pong for b-features
    float* aggA   = ws;  ws += na;
    float* aggB1  = ws;  ws += nb;
    float* aggB2  = ws;  ws += nb;
    float* cnt_aa = ws;  ws += kNA;
    float* cnt_ab = ws;  ws += kNB;
    float* cnt_bb = ws;  // += kNB

    // degree counts (constant across gates/layers): cnt_aa, cnt_ab, cnt_bb contiguous
    hipMemsetAsync(cnt_aa, 0, (size_t)(kNA + 2 * kNB) * sizeof(float), stream);
    count_kernel<<<(E_aa + 255) / 256, 256, 0, stream>>>(cnt_aa, dst_aa, E_aa);
    count_kernel<<<(E_ab + 255) / 256, 256, 0, stream>>>(cnt_ab, dst_ab, E_ab);
    count_kernel<<<(E_bb + 255) / 256, 256, 0, stream>>>(cnt_bb, dst_bb, E_bb);

    const unsigned gs_aa = (unsigned)(((size_t)E_aa * 16 + 255) / 256);
    const unsigned gs_ab = (unsigned)(((size_t)E_ab * 16 + 255) / 256);
    const unsigned gs_bb = (unsigned)(((size_t)E_bb * 16 + 255) / 256);

    for (int g = 0; g < 4; ++g) {
        float* tA = tmpA;
        float* tB = tmpB;
        // lin projections (no bias here; lin_b is added after conv layers, per reference)
        launch_gemm(stream, tA, x_a, nullptr, lin_w + (size_t)(g * 2 + 0) * kFIN * kC, nullptr, kNA, kFIN, 0);
        launch_gemm(stream, tB, x_b, nullptr, lin_w + (size_t)(g * 2 + 1) * kFIN * kC, nullptr, kNB, kFIN, 0);

        for (int l = 0; l < 2; ++l) {
            float* outA = (l == 1) ? (gate_a + (size_t)g * na) : (tmpA + na);
            float* outB = (l == 1) ? (gate_b + (size_t)g * nb) : (tmpB + nb);
            const float* Wl = conv_wl + (size_t)((g * 2 + l) * 3) * kC * kC;
            const float* Wr = conv_wr + (size_t)((g * 2 + l) * 3) * kC * kC;
            const float* Bb = conv_b  + (size_t)((g * 2 + l) * 3) * kC;

            // a -> a
            hipMemsetAsync(aggA, 0, na * sizeof(float), stream);
            scatter_kernel<<<gs_aa, 256, 0, stream>>>(aggA, tA, src_aa, dst_aa, E_aa);
            launch_gemm(stream, outA, aggA, cnt_aa, Wl + 0 * kC * kC, Bb + 0 * kC, kNA, kC, 0);
            launch_gemm(stream, outA, tA,   nullptr, Wr + 0 * kC * kC, nullptr,    kNA, kC, 1);

            // a -> b  and  b -> b (HeteroConv sums into dst b)
            hipMemsetAsync(aggB1, 0, nb * sizeof(float), stream);
            hipMemsetAsync(aggB2, 0, nb * sizeof(float), stream);
            scatter_kernel<<<gs_ab, 256, 0, stream>>>(aggB1, tA, src_ab, dst_ab, E_ab);
            scatter_kernel<<<gs_bb, 256, 0, stream>>>(aggB2, tB, src_bb, dst_bb, E_bb);
            launch_gemm(stream, outB, aggB1, cnt_ab, Wl + 1 * kC * kC, Bb + 1 * kC, kNB, kC, 0);
            launch_gemm(stream, outB, tB,    nullptr, Wr + 1 * kC * kC, nullptr,    kNB, kC, 1);
            launch_gemm(stream, outB, aggB2, cnt_bb, Wl + 2 * kC * kC, Bb + 2 * kC, kNB, kC, 1);
            launch_gemm(stream, outB, tB,    nullptr, Wr + 2 * kC * kC, nullptr,    kNB, kC, 1);

            tA = outA;
            tB = outB;
        }
    }

    // outputs: (h_a, h_b, cn_a, cn_b) concatenated flat
    float* out  = (float*)d_out;
    float* h_a  = out;
    float* h_b  = out + na;
    float* cn_a = out + na + nb;
    float* cn_b = out + na + nb + na;

    lstm_kernel<<<(unsigned)((na + 255) / 256), 256, 0, stream>>>(
        h_a, cn_a,
        gate_a + 0 * na, gate_a + 1 * na, gate_a + 2 * na, gate_a + 3 * na,
        c_a, lin_b, /*t=*/0, (long long)na);
    lstm_kernel<<<(unsigned)((nb + 255) / 256), 256, 0, stream>>>(
        h_b, cn_b,
        gate_b + 0 * nb, gate_b + 1 * nb, gate_b + 2 * nb, gate_b + 3 * nb,
        c_b, lin_b, /*t=*/1, (long long)nb);
}